// ContextAwarePathIntegrator_42743514529894
// MI455X (gfx1250) — compile-verified
//
#include <hip/hip_runtime.h>
#include <math.h>

typedef __attribute__((ext_vector_type(2))) float v2f;
typedef __attribute__((ext_vector_type(8))) float v8f;

#define PI_F 3.14159265358979323846f

__device__ __forceinline__ v8f wmma4(v2f a, v2f b, v8f c) {
    // V_WMMA_F32_16X16X4_F32 : D = A(16x4,f32) * B(4x16,f32) + C(16x16,f32)
    return __builtin_amdgcn_wmma_f32_16x16x4_f32(false, a, false, b, (short)0, c,
                                                 false, false);
}

// single-hop cross-lane gather (ds_bpermute_b32): dst[lane] = src[byteIdx/4]
__device__ __forceinline__ float bperm(int byteIdx, float x) {
    return __int_as_float(__builtin_amdgcn_ds_bpermute(byteIdx, __float_as_int(x)));
}

// Branch-free fp32 sincos: quadrant reduction + Cody-Waite pi/2 + minimax
// polys on [-pi/4, pi/4]. Full fp32 accuracy for |x| <~ 1e4.
__device__ __forceinline__ void fast_sincosf(float x, float& s, float& c) {
    const float k = rintf(x * 0.63661977236758134f);   // x * 2/pi
    const int   q = (int)k;
    float r = fmaf(k, -1.57079637050628662e+0f, x);
    r       = fmaf(k,  4.37113882867379188e-8f, r);
    const float r2 = r * r;
    float sp = fmaf(r2, 2.7525562e-6f, -1.9840874e-4f);
    sp = fmaf(r2, sp, 8.3333310e-3f);
    sp = fmaf(r2, sp, -1.6666667e-1f);
    sp = fmaf(r * r2, sp, r);
    float cp = fmaf(r2, 2.4760495e-5f, -1.3888378e-3f);
    cp = fmaf(r2, cp, 4.1666638e-2f);
    cp = fmaf(r2, cp, -5.0e-1f);
    cp = fmaf(r2, cp, 1.0f);
    const bool swap = (q & 1) != 0;
    float ss = swap ? cp : sp;
    float cc = swap ? sp : cp;
    if ((q & 2) != 0)       ss = -ss;
    if (((q + 1) & 2) != 0) cc = -cc;
    s = ss;
    c = cc;
}

struct PoseAct { float s, c, dx, dy; };

// State-independent per-step work: deltaT = WaT @ actionT (4 WMMAs),
// broadcast (drot,dx,dy) from lo lanes, sincos. Software-pipelined one
// step ahead of the recurrence.
__device__ __forceinline__ PoseAct compute_pa(const float* ap, int u,
                                              const v2f* aWa, v8f accD0,
                                              int hi, int idxLo) {
    v8f d = accD0;
#pragma unroll
    for (int j = 0; j < 4; ++j) {
        // B operand (actionT): lane row n, k = 4j + 2*hi (+1); 8B-aligned
        v2f bAct = *(const v2f*)(ap + u * 16 + 4 * j + 2 * hi);
        d = wmma4(aWa[j], bAct, d);
    }
    // deltaT D-layout: (out-col c, row m) -> lane m, vgpr c (c<3, lo half)
    const float d0 = bperm(idxLo, d[0]);
    const float d1 = bperm(idxLo, d[1]);
    const float d2 = bperm(idxLo, d[2]);
    PoseAct r;
    fast_sincosf(d0 * PI_F, r.s, r.c);
    r.dx = d1;
    r.dy = d2;
    return r;
}

__global__ __launch_bounds__(256)
void cap_path_integrate(const float* __restrict__ t_init,
                        const float* __restrict__ action,
                        const float* __restrict__ t_direct,
                        const unsigned char* __restrict__ mask,
                        const float* __restrict__ Wa,
                        const float* __restrict__ ba,
                        const float* __restrict__ W1,
                        const float* __restrict__ b1,
                        const float* __restrict__ W2,
                        const float* __restrict__ b2,
                        float* __restrict__ out,
                        int B, int T)
{
    const int tid  = threadIdx.x;
    const int wv   = tid >> 5;
    const int lane = tid & 31;
    const int n    = lane & 15;      // matrix row/col index owned by this lane
    const int hi   = lane >> 4;      // half-wave

    const int idxLo = n * 4;          // gather from lane n   (bytes)
    const int idxHi = (n + 16) * 4;   // gather from lane n+16

    const int b = (blockIdx.x * 8 + wv) * 16 + n;   // batch row (replicated in halves)

    // ---- static A-operands (transposed weights), WMMA A-layout ----
    // A layout (16x4 f32): lane l -> M = l%16, vgpr v -> K = 2*hi + v
    v2f aWa[4];                       // WaT (3x16), M padded to 16
#pragma unroll
    for (int j = 0; j < 4; ++j) {
        const int k0 = 4 * j + 2 * hi;
        aWa[j].x = (n < 3) ? Wa[k0 * 3 + n]       : 0.0f;
        aWa[j].y = (n < 3) ? Wa[(k0 + 1) * 3 + n] : 0.0f;
    }
    v2f aW1[2][3];                    // W1T (32x12): 2 M-tiles x 3 K-chunks
#pragma unroll
    for (int tt = 0; tt < 2; ++tt)
#pragma unroll
        for (int j = 0; j < 3; ++j) {
            const int k0 = 4 * j + 2 * hi;
            aW1[tt][j].x = W1[k0 * 32 + 16 * tt + n];
            aW1[tt][j].y = W1[(k0 + 1) * 32 + 16 * tt + n];
        }
    v2f aW2[8];                       // W2T (3x32), M padded to 16
#pragma unroll
    for (int j = 0; j < 8; ++j) {
        const int k0 = 4 * j + 2 * hi;
        aW2[j].x = (n < 3) ? W2[k0 * 3 + n]       : 0.0f;
        aW2[j].y = (n < 3) ? W2[(k0 + 1) * 3 + n] : 0.0f;
    }
    // ---- bias inits along the M dimension: vgpr v holds M = v + 8*hi ----
    v8f accD0, accO0, accH0[2];
#pragma unroll
    for (int v = 0; v < 8; ++v) {
        accD0[v] = (hi == 0 && v < 3) ? ba[v] : 0.0f;
        accO0[v] = (hi == 0 && v < 3) ? b2[v] : 0.0f;
        accH0[0][v] = b1[v + 8 * hi];
        accH0[1][v] = b1[16 + v + 8 * hi];
    }

    // Recurrent 3x3 state in fp32 registers (replicated in both halves).
    float tc[9];
#pragma unroll
    for (int i = 0; i < 9; ++i) tc[i] = t_init[(size_t)b * 9 + i];

    const float* ap  = action   + (size_t)b * T * 16;
    const float* tdp = t_direct + (size_t)b * T * 9;
    const unsigned char* mkp = mask + (size_t)b * T;
    float* op = out + (size_t)b * T * 9;

    // prologue: pose_act for t = 0
    PoseAct pa = compute_pa(ap, 0, aWa, accD0, hi, idxLo);

    for (int t = 0; t < T; ++t) {
        // hoisted control inputs: latency hides under the WMMA chain
        const bool mk = mkp[t] != 0;
        float td[9];
#pragma unroll
        for (int i = 0; i < 9; ++i) td[i] = tdp[t * 9 + i];

        // kick off next step's state-independent pose_act (interleaves freely)
        const int tn = (t + 1 < T) ? (t + 1) : t;
        PoseAct paN = compute_pa(ap, tn, aWa, accD0, hi, idxLo);

        // ---------------- state-dependent chain for step t ----------------
        float pose[12];
        pose[0] = tc[0]; pose[1]  = tc[1];  pose[2]  = tc[2];
        pose[3] = tc[3]; pose[4]  = tc[4];  pose[5]  = tc[5];
        pose[6] = pa.c;  pose[7]  = -pa.s;  pose[8]  = pa.dx;
        pose[9] = pa.s;  pose[10] = pa.c;   pose[11] = pa.dy;

        // hT = W1T @ poseT  (6 WMMAs) ; B operand = per-lane pose selects
        v8f hT0 = accH0[0];
        v8f hT1 = accH0[1];
#pragma unroll
        for (int j = 0; j < 3; ++j) {
            v2f bP;
            bP.x = hi ? pose[4 * j + 2] : pose[4 * j + 0];
            bP.y = hi ? pose[4 * j + 3] : pose[4 * j + 1];
            hT0 = wmma4(aW1[0][j], bP, hT0);
            hT1 = wmma4(aW1[1][j], bP, hT1);
        }
#pragma unroll
        for (int v = 0; v < 8; ++v) {
            hT0[v] = fmaxf(hT0[v], 0.0f);
            hT1[v] = fmaxf(hT1[v], 0.0f);
        }

        // out3T = W2T @ hT  (8 WMMAs); B chunks gathered from hT D-layout
        // via independent single-hop ds_bpermute (no LDS round trip).
        v8f o = accO0;
#pragma unroll
        for (int j = 0; j < 8; ++j) {
            const v8f& tile = (j < 4) ? hT0 : hT1;
            const int jj   = j & 3;
            const int base = (jj & 1) * 4;          // source vgpr group
            const int gidx = (jj >= 2) ? idxHi : idxLo;  // source half
            const float s0 = hi ? tile[base + 2] : tile[base + 0];
            const float s1 = hi ? tile[base + 3] : tile[base + 1];
            v2f bH;
            bH.x = bperm(gidx, s0);
            bH.y = bperm(gidx, s1);
            o = wmma4(aW2[j], bH, o);
        }
        // out3T D-layout: (rot,tx,ty) for row m at lane m, vgprs 0..2
        const float rot2 = bperm(idxLo, o[0]);
        const float tx   = bperm(idxLo, o[1]);
        const float ty   = bperm(idxLo, o[2]);
        float s2, c2;
        fast_sincosf(rot2, s2, c2);

        // t_out = t_ctx @ [[c,-s,tx],[s,c,ty],[0,0,1]]
        float to[9];
#pragma unroll
        for (int i = 0; i < 3; ++i) {
            const float r0 = tc[i * 3 + 0], r1 = tc[i * 3 + 1], r2 = tc[i * 3 + 2];
            to[i * 3 + 0] = fmaf(r0, c2, r1 * s2);
            to[i * 3 + 1] = fmaf(r1, c2, -r0 * s2);
            to[i * 3 + 2] = fmaf(r0, tx, fmaf(r1, ty, r2));
        }
        // split the 9 output floats across the two half-waves
        if (hi == 0) {
            op[t * 9 + 0] = to[0]; op[t * 9 + 1] = to[1]; op[t * 9 + 2] = to[2];
            op[t * 9 + 3] = to[3]; op[t * 9 + 4] = to[4];
        } else {
            op[t * 9 + 5] = to[5]; op[t * 9 + 6] = to[6];
            op[t * 9 + 7] = to[7]; op[t * 9 + 8] = to[8];
        }
        // masked reset (pure selects; loads already in flight)
#pragma unroll
        for (int i = 0; i < 9; ++i) tc[i] = mk ? td[i] : to[i];

        pa = paN;
    }
}

extern "C" void kernel_launch(void* const* d_in, const int* in_sizes, int n_in,
                              void* d_out, int out_size, void* d_ws, size_t ws_size,
                              hipStream_t stream) {
    (void)n_in; (void)out_size; (void)d_ws; (void)ws_size;
    const float* t_init   = (const float*)d_in[0];
    const float* action   = (const float*)d_in[1];
    const float* t_direct = (const float*)d_in[2];
    const unsigned char* mask = (const unsigned char*)d_in[3];
    const float* Wa = (const float*)d_in[4];
    const float* ba = (const float*)d_in[5];
    const float* W1 = (const float*)d_in[6];
    const float* b1 = (const float*)d_in[7];
    const float* W2 = (const float*)d_in[8];
    const float* b2 = (const float*)d_in[9];
    float* out = (float*)d_out;

    const int B = in_sizes[0] / 9;          // t_init: (B,1,3,3) -> 8192
    const int T = in_sizes[1] / (B * 16);   // action: (B,T,16)  -> 200
    const int waves  = B / 16;              // 16 batch rows per wave
    const int blocks = waves / 8;           // 8 waves (256 threads) per block
    cap_path_integrate<<<blocks, 256, 0, stream>>>(
        t_init, action, t_direct, mask, Wa, ba, W1, b1, W2, b2, out, B, T);
}